// GraphSAGE_53815940219287
// MI455X (gfx1250) — compile-verified
//
#include <hip/hip_runtime.h>
#include <hip/hip_bf16.h>
#include <math.h>

#define N_NODES   100000
#define N_EDGES   1600000
#define HID       128
#define OUT_DIM   64
#define N_GRAPHS  128

// pair-rows (k/2) and padded pitch for conflict-free ds_load_b64
#define KPAIRS    64
#define PITCH2    144   // 128 cols + 16 pad (float2 units); 288 dwords ≡ 32 (mod 64)

typedef __attribute__((ext_vector_type(2))) float v2f;
typedef __attribute__((ext_vector_type(8))) float v8f;

// ---------------------------------------------------------------------------
// dtype probe: JAX may deliver edge_index/batch as int32 (x64 disabled) or
// int64. True int64 indices are < 2^31, so any nonzero high dword when
// reading as int64 means the buffer is really packed int32.
// ---------------------------------------------------------------------------
__global__ void detect_idx_kernel(const void* __restrict__ ei, int* __restrict__ flag) {
    if (blockIdx.x == 0 && threadIdx.x == 0) {
        const unsigned long long* p = (const unsigned long long*)ei;
        int is32 = 0;
        for (int i = 0; i < 256; ++i) {
            if ((p[i] >> 32) != 0ull) is32 = 1;
        }
        *flag = is32;
    }
}

__device__ __forceinline__ int load_idx(const void* base, long long i, int is32) {
    return is32 ? ((const int*)base)[i] : (int)((const long long*)base)[i];
}

// ---------------------------------------------------------------------------
__global__ void zero_f32_kernel(float* __restrict__ p, int n) {
    int i = blockIdx.x * blockDim.x + threadIdx.x;
    int stride = gridDim.x * blockDim.x;
    for (; i < n; i += stride) p[i] = 0.0f;
}

// ---------------------------------------------------------------------------
// Edge scatter-add: one edge per 32 lanes, float4 per lane, f32 atomics.
// L2-resident (node features = 51.2 MB < 192 MB L2).
// ---------------------------------------------------------------------------
__global__ __launch_bounds__(256)
void scatter_add_kernel(const float* __restrict__ h, float* __restrict__ agg,
                        const void* __restrict__ ei, const int* __restrict__ flag) {
    long long tid = (long long)blockIdx.x * blockDim.x + threadIdx.x;
    int edge = (int)(tid >> 5);
    int lane = (int)(tid & 31);
    if (edge >= N_EDGES) return;
    int is32 = *flag;
    int s = load_idx(ei, edge, is32);
    int d = load_idx(ei, (long long)N_EDGES + edge, is32);
    float4 v = ((const float4*)(h + (size_t)s * HID))[lane];
    float* out = agg + (size_t)d * HID + lane * 4;
    atomicAdd(out + 0, v.x);
    atomicAdd(out + 1, v.y);
    atomicAdd(out + 2, v.z);
    atomicAdd(out + 3, v.w);
}

// ---------------------------------------------------------------------------
// Fused SAGE layer GEMM:  agg <- relu(agg @ Wl + h @ Wr + bias)   (in place)
// Block: 128 threads (4 waves). Wave computes 16 rows x 128 cols with
// v_wmma_f32_16x16x4_f32 (exact fp32 path).
// Weights staged in LDS pair-interleaved: sW[k2][n] = {W[2k2][n], W[2k2+1][n]}
// so each B fragment is ONE aligned ds_load_b64 (conflict-free via pitch pad).
// A frag (16x4 f32): lane l: M = l&15, K = 4*kt + 2*(l>>4) + v
// B frag (4x16 f32): lane l: N = l&15, K = 4*kt + 2*(l>>4) + v
// C/D  (16x16 f32):  reg r: M = r + 8*(l>>4), N = l&15
// ---------------------------------------------------------------------------
__global__ __launch_bounds__(128)
void sage_gemm_kernel(float* __restrict__ agg,        // [N_NODES][HID] in/out
                      const float* __restrict__ h,    // [N_NODES][HID]
                      const float* __restrict__ Wl,   // [HID][HID] (K-major rows)
                      const float* __restrict__ Wr,   // [HID][HID]
                      const float* __restrict__ bias) // [HID]
{
    __shared__ v2f sWl2[KPAIRS * PITCH2];   // 72 KB
    __shared__ v2f sWr2[KPAIRS * PITCH2];   // 72 KB

    // stage + pair-interleave both weight matrices
    {
        const float4* wl4 = (const float4*)Wl;
        const float4* wr4 = (const float4*)Wr;
        for (int i = threadIdx.x; i < KPAIRS * 32; i += 128) {
            const int k2 = i >> 5;        // pair-row
            const int ng = i & 31;        // group of 4 columns
            float4 l0 = wl4[(2 * k2 + 0) * 32 + ng];
            float4 l1 = wl4[(2 * k2 + 1) * 32 + ng];
            float4 r0 = wr4[(2 * k2 + 0) * 32 + ng];
            float4 r1 = wr4[(2 * k2 + 1) * 32 + ng];
            v2f* dl = &sWl2[k2 * PITCH2 + ng * 4];
            v2f* dr = &sWr2[k2 * PITCH2 + ng * 4];
            dl[0] = v2f{l0.x, l1.x}; dl[1] = v2f{l0.y, l1.y};
            dl[2] = v2f{l0.z, l1.z}; dl[3] = v2f{l0.w, l1.w};
            dr[0] = v2f{r0.x, r1.x}; dr[1] = v2f{r0.y, r1.y};
            dr[2] = v2f{r0.z, r1.z}; dr[3] = v2f{r0.w, r1.w};
        }
    }
    __syncthreads();

    const int lane  = threadIdx.x & 31;
    const int wave  = threadIdx.x >> 5;
    const int lhalf = lane >> 4;   // 0 or 1
    const int l15   = lane & 15;

    const int r0 = blockIdx.x * 64 + wave * 16;
    int arow = r0 + l15;
    if (arow >= N_NODES) arow = N_NODES - 1;  // clamp loads; stores are guarded
    const float* aRow = agg + (size_t)arow * HID;
    const float* hRow = h   + (size_t)arow * HID;

    v8f acc[8];
#pragma unroll
    for (int nt = 0; nt < 8; ++nt) acc[nt] = {};

    for (int kt = 0; kt < 32; ++kt) {
        const int k = kt * 4 + 2 * lhalf;            // this lane's K base
        const int kbase = (kt * 2 + lhalf) * PITCH2 + l15;  // LDS pair-row base
        v2f aA = *(const v2f*)&aRow[k];              // global_load_b64
        v2f aH = *(const v2f*)&hRow[k];
#pragma unroll
        for (int nt = 0; nt < 8; ++nt) {
            v2f bL = sWl2[kbase + nt * 16];          // one ds_load_b64
            acc[nt] = __builtin_amdgcn_wmma_f32_16x16x4_f32(
                false, aA, false, bL, (short)0, acc[nt], false, false);
            v2f bR = sWr2[kbase + nt * 16];
            acc[nt] = __builtin_amdgcn_wmma_f32_16x16x4_f32(
                false, aH, false, bR, (short)0, acc[nt], false, false);
        }
    }

    // bias + relu + in-place store (each row owned by exactly one wave; all
    // reads of these rows completed before stores issue — in-order wave)
#pragma unroll
    for (int nt = 0; nt < 8; ++nt) {
        const int col = nt * 16 + l15;
        const float b = bias[col];
#pragma unroll
        for (int r = 0; r < 8; ++r) {
            const int row = r0 + r + 8 * lhalf;
            if (row < N_NODES) {
                float v = acc[nt][r] + b;
                agg[(size_t)row * HID + col] = fmaxf(v, 0.0f);
            }
        }
    }
}

// ---------------------------------------------------------------------------
// Global mean-pool (atomic segment sum): node per 32 lanes, float4 per lane.
// ---------------------------------------------------------------------------
__global__ __launch_bounds__(256)
void pool_kernel(const float* __restrict__ h, const void* __restrict__ batch,
                 float* __restrict__ sums, float* __restrict__ counts,
                 const int* __restrict__ flag) {
    long long tid = (long long)blockIdx.x * blockDim.x + threadIdx.x;
    int node = (int)(tid >> 5);
    int lane = (int)(tid & 31);
    if (node >= N_NODES) return;
    int is32 = *flag;
    int g = load_idx(batch, node, is32);
    float4 v = ((const float4*)(h + (size_t)node * HID))[lane];
    float* o = sums + (size_t)g * HID + lane * 4;
    atomicAdd(o + 0, v.x);
    atomicAdd(o + 1, v.y);
    atomicAdd(o + 2, v.z);
    atomicAdd(o + 3, v.w);
    if (lane == 0) atomicAdd(counts + g, 1.0f);
}

// ---------------------------------------------------------------------------
// FC + log_softmax: one block per graph, 64 threads (one per output class).
// ---------------------------------------------------------------------------
__global__ __launch_bounds__(64)
void fc_logsoftmax_kernel(const float* __restrict__ sums,
                          const float* __restrict__ counts,
                          const float* __restrict__ Wfc,   // [HID][OUT_DIM]
                          const float* __restrict__ bfc,   // [OUT_DIM]
                          float* __restrict__ out) {       // [N_GRAPHS][OUT_DIM]
    __shared__ float pooled[HID];
    __shared__ float red[OUT_DIM];
    const int g = blockIdx.x;
    const int o = threadIdx.x;

    const float inv = 1.0f / fmaxf(counts[g], 1.0f);
    for (int k = o; k < HID; k += OUT_DIM) pooled[k] = sums[(size_t)g * HID + k] * inv;
    __syncthreads();

    float acc = bfc[o];
    for (int k = 0; k < HID; ++k) acc = fmaf(pooled[k], Wfc[k * OUT_DIM + o], acc);

    red[o] = acc;
    __syncthreads();
    for (int s = OUT_DIM / 2; s > 0; s >>= 1) {
        if (o < s) red[o] = fmaxf(red[o], red[o + s]);
        __syncthreads();
    }
    const float m = red[0];
    __syncthreads();
    red[o] = __expf(acc - m);
    __syncthreads();
    for (int s = OUT_DIM / 2; s > 0; s >>= 1) {
        if (o < s) red[o] += red[o + s];
        __syncthreads();
    }
    const float lse = m + logf(red[0]);
    out[(size_t)g * OUT_DIM + o] = acc - lse;
}

// ---------------------------------------------------------------------------
extern "C" void kernel_launch(void* const* d_in, const int* in_sizes, int n_in,
                              void* d_out, int out_size, void* d_ws, size_t ws_size,
                              hipStream_t stream) {
    (void)in_sizes; (void)n_in; (void)out_size; (void)ws_size;

    const float* x     = (const float*)d_in[0];
    const void*  ei    = d_in[1];              // (2, N_EDGES) int64 or int32
    const void*  batch = d_in[2];              // (N_NODES,)   int64 or int32
    const float* Wl0 = (const float*)d_in[3];
    const float* bl0 = (const float*)d_in[4];
    const float* Wr0 = (const float*)d_in[5];
    const float* Wl1 = (const float*)d_in[6];
    const float* bl1 = (const float*)d_in[7];
    const float* Wr1 = (const float*)d_in[8];
    const float* Wl2 = (const float*)d_in[9];
    const float* bl2 = (const float*)d_in[10];
    const float* Wr2 = (const float*)d_in[11];
    const float* Wfc = (const float*)d_in[12];
    const float* bfc = (const float*)d_in[13];
    float* out = (float*)d_out;

    const size_t featBytes = (size_t)N_NODES * HID * sizeof(float);
    char* ws = (char*)d_ws;
    float* bufA   = (float*)ws;                               // 51.2 MB
    float* bufB   = (float*)(ws + featBytes);                 // 51.2 MB
    float* sums   = (float*)(ws + 2 * featBytes);             // 64 KB
    float* counts = sums + (size_t)N_GRAPHS * HID;            // 512 B
    int*   flag   = (int*)(counts + N_GRAPHS);

    const int feat = N_NODES * HID;
    const dim3 zGrid(4096), zBlk(256);
    const int scatterBlocks = (N_EDGES * 32 + 255) / 256;   // 200000
    const int gemmBlocks    = (N_NODES + 63) / 64;          // 1563
    const int poolBlocks    = (N_NODES * 32 + 255) / 256;   // 12500

    detect_idx_kernel<<<1, 32, 0, stream>>>(ei, flag);

    // ---- layer 1: h1 = relu(agg(x)@Wl0 + x@Wr0 + b0), stored in bufA
    zero_f32_kernel<<<zGrid, zBlk, 0, stream>>>(bufA, feat);
    scatter_add_kernel<<<scatterBlocks, 256, 0, stream>>>(x, bufA, ei, flag);
    sage_gemm_kernel<<<gemmBlocks, 128, 0, stream>>>(bufA, x, Wl0, Wr0, bl0);

    // ---- layer 2: h2 in bufB
    zero_f32_kernel<<<zGrid, zBlk, 0, stream>>>(bufB, feat);
    scatter_add_kernel<<<scatterBlocks, 256, 0, stream>>>(bufA, bufB, ei, flag);
    sage_gemm_kernel<<<gemmBlocks, 128, 0, stream>>>(bufB, bufA, Wl1, Wr1, bl1);

    // ---- layer 3: h3 in bufA
    zero_f32_kernel<<<zGrid, zBlk, 0, stream>>>(bufA, feat);
    scatter_add_kernel<<<scatterBlocks, 256, 0, stream>>>(bufB, bufA, ei, flag);
    sage_gemm_kernel<<<gemmBlocks, 128, 0, stream>>>(bufA, bufB, Wl2, Wr2, bl2);

    // ---- mean pool + FC + log_softmax
    zero_f32_kernel<<<16, 256, 0, stream>>>(sums, N_GRAPHS * HID + N_GRAPHS);
    pool_kernel<<<poolBlocks, 256, 0, stream>>>(bufA, batch, sums, counts, flag);
    fc_logsoftmax_kernel<<<N_GRAPHS, OUT_DIM, 0, stream>>>(sums, counts, Wfc, bfc, out);
}